// GraphTransformerLayer_28613072126576
// MI455X (gfx1250) — compile-verified
//
#include <hip/hip_runtime.h>
#include <hip/hip_bf16.h>
#include <math.h>

typedef float v2f __attribute__((ext_vector_type(2)));
typedef float v8f __attribute__((ext_vector_type(8)));

// D = A(16x4 f32) * B(4x16 f32) + C(16x16 f32), wave32.
// A layout: lanes 0-15 -> M=lane, v0=K(k0), v1=K(k0+1); lanes 16-31 -> K(k0+2),K(k0+3)
// B layout (mirror): lanes 0-15 -> N=lane, v0=row k0, v1=row k0+1; lanes 16-31 -> rows k0+2,k0+3
// C/D layout: c[r] -> row (r + 8*(lane>=16)), col (lane&15)
#define WMMA_F32(cacc, av, bv) \
  (cacc) = __builtin_amdgcn_wmma_f32_16x16x4_f32(false, (av), false, (bv), (short)0, (cacc), false, false)

// ---------------------------------------------------------------------------
// Q/K/V projection: 16-row block per workgroup, 4 waves = 4 column tiles of 16.
// ---------------------------------------------------------------------------
__global__ void qkv_kernel(const float* __restrict__ h,
                           const float* __restrict__ Wq, const float* __restrict__ bq,
                           const float* __restrict__ Wk, const float* __restrict__ bk,
                           const float* __restrict__ Wv, const float* __restrict__ bv,
                           float* __restrict__ Q, float* __restrict__ K, float* __restrict__ V,
                           int N)
{
    __shared__ __align__(16) float As[16 * 68];         // 272B row stride: aligned + conflict-free
    const int row0 = blockIdx.x * 16;
    const bool full = (row0 + 16 <= N);

    if (full) {                                         // vector fill, no guards
        for (int i = threadIdx.x; i < 256; i += blockDim.x) {
            int r = i >> 4, c = (i & 15) * 4;
            *(float4*)(As + r * 68 + c) = *(const float4*)(h + (size_t)(row0 + r) * 64 + c);
        }
    } else {
        for (int i = threadIdx.x; i < 16 * 64; i += blockDim.x) {
            int r = i >> 6, c = i & 63;
            int gr = row0 + r;
            As[r * 68 + c] = (gr < N) ? h[(size_t)gr * 64 + c] : 0.f;
        }
    }
    __syncthreads();

    const int lane = threadIdx.x & 31;
    const int wv   = threadIdx.x >> 5;                  // 0..3 -> column tile
    const int hi   = lane >> 4;
    const int lo   = lane & 15;
    const int n0   = wv * 16;

#define QKV_ONE(W, BIAS, OUT)                                                   \
    do {                                                                        \
        v8f c = {};                                                             \
        _Pragma("unroll")                                                       \
        for (int k0 = 0; k0 < 64; k0 += 4) {                                    \
            int kb = k0 + 2 * hi;                                               \
            v2f a = { As[lo * 68 + kb], As[lo * 68 + kb + 1] };                 \
            v2f b = { W[kb * 64 + n0 + lo], W[(kb + 1) * 64 + n0 + lo] };       \
            WMMA_F32(c, a, b);                                                  \
        }                                                                       \
        float bias = BIAS[n0 + lo];                                             \
        if (full) {                                                             \
            _Pragma("unroll")                                                   \
            for (int r = 0; r < 8; ++r)                                         \
                OUT[(size_t)(row0 + r + 8 * hi) * 64 + n0 + lo] = c[r] + bias;  \
        } else {                                                                \
            _Pragma("unroll")                                                   \
            for (int r = 0; r < 8; ++r) {                                       \
                int gr = row0 + r + 8 * hi;                                     \
                if (gr < N) OUT[(size_t)gr * 64 + n0 + lo] = c[r] + bias;       \
            }                                                                   \
        }                                                                       \
    } while (0)

    QKV_ONE(Wq, bq, Q);
    QKV_ONE(Wk, bk, K);
    QKV_ONE(Wv, bv, V);
#undef QKV_ONE
}

// ---------------------------------------------------------------------------
// Edge attention: two edges per wave (16 lanes each), 4 channels per lane so all
// gathers are b128. Per-head dot needs a single shfl_xor; fp32 L2 atomics.
// ---------------------------------------------------------------------------
__global__ void edge_attn(const float* __restrict__ Q, const float* __restrict__ K,
                          const float* __restrict__ V,
                          const int* __restrict__ src, const int* __restrict__ dst,
                          float* __restrict__ wV, float* __restrict__ z, int E)
{
    const float inv_scale = 0.3535533905932738f;        // 1/sqrt(DH), DH=8
    const int lane = threadIdx.x & 31;
    const int half = lane >> 4;                         // which edge of the pair
    const int l    = lane & 15;
    const int c4   = l * 4;                             // 4 channels per lane (one head per 2 lanes)
    const int head = l >> 1;
    const int wid  = (blockIdx.x * blockDim.x + threadIdx.x) >> 5;
    const int nw   = (gridDim.x * blockDim.x) >> 5;

    for (int e0 = wid * 2; e0 < E; e0 += nw * 2) {
        int e = e0 + half;
        bool valid = (e < E);
        int ee = valid ? e : (E - 1);                   // clamp: loads stay in-bounds
        int s = src[ee];
        int d = dst[ee];
        float4 kv = *(const float4*)(K + (size_t)s * 64 + c4);
        float4 qv = *(const float4*)(Q + (size_t)d * 64 + c4);
        float p = kv.x * qv.x + kv.y * qv.y + kv.z * qv.z + kv.w * qv.w;
        p += __shfl_xor(p, 1, 32);                      // 8-ch head = 2 lanes
        float sc = expf(fminf(fmaxf(p * inv_scale, -5.f), 5.f));
        float4 vv = *(const float4*)(V + (size_t)s * 64 + c4);
        if (valid) {
            atomicAdd(&wV[(size_t)d * 64 + c4 + 0], vv.x * sc);
            atomicAdd(&wV[(size_t)d * 64 + c4 + 1], vv.y * sc);
            atomicAdd(&wV[(size_t)d * 64 + c4 + 2], vv.z * sc);
            atomicAdd(&wV[(size_t)d * 64 + c4 + 3], vv.w * sc);
            if ((l & 1) == 0) atomicAdd(&z[(size_t)d * 8 + head], sc);
        }
    }
}

// ---------------------------------------------------------------------------
// x1 = h + wV/z (0 where z==0) in float4, plus per-channel sum/sumsq for BN1.
// Grid stride is a multiple of 16 float4s so each thread owns 4 fixed channels.
// ---------------------------------------------------------------------------
__global__ void attn_resid_stats(const float* __restrict__ h, const float* __restrict__ wV,
                                 const float* __restrict__ z, float* __restrict__ x1,
                                 float* __restrict__ sum1, float* __restrict__ sq1, int N)
{
    __shared__ float ssum[64], ssq[64];
    if (threadIdx.x < 64) { ssum[threadIdx.x] = 0.f; ssq[threadIdx.x] = 0.f; }
    __syncthreads();
    const int cg = (threadIdx.x & 15) * 4;              // channels cg..cg+3 (single head)
    const int hd = cg >> 3;
    float ls0 = 0, ls1 = 0, ls2 = 0, ls3 = 0;
    float lq0 = 0, lq1 = 0, lq2 = 0, lq3 = 0;
    const int total4 = N * 16;
    const int stride = gridDim.x * blockDim.x;          // multiple of 16
    for (int i = blockIdx.x * blockDim.x + threadIdx.x; i < total4; i += stride) {
        int node = i >> 4;
        float zz  = z[(size_t)node * 8 + hd];
        float inv = (zz == 0.f) ? 0.f : 1.f / zz;
        float4 w  = *(const float4*)(wV + (size_t)i * 4);
        float4 hv = *(const float4*)(h  + (size_t)i * 4);
        float4 v;
        v.x = fmaf(w.x, inv, hv.x);
        v.y = fmaf(w.y, inv, hv.y);
        v.z = fmaf(w.z, inv, hv.z);
        v.w = fmaf(w.w, inv, hv.w);
        *(float4*)(x1 + (size_t)i * 4) = v;
        ls0 += v.x; lq0 += v.x * v.x;
        ls1 += v.y; lq1 += v.y * v.y;
        ls2 += v.z; lq2 += v.z * v.z;
        ls3 += v.w; lq3 += v.w * v.w;
    }
    atomicAdd(&ssum[cg + 0], ls0); atomicAdd(&ssq[cg + 0], lq0);
    atomicAdd(&ssum[cg + 1], ls1); atomicAdd(&ssq[cg + 1], lq1);
    atomicAdd(&ssum[cg + 2], ls2); atomicAdd(&ssq[cg + 2], lq2);
    atomicAdd(&ssum[cg + 3], ls3); atomicAdd(&ssq[cg + 3], lq3);
    __syncthreads();
    if (threadIdx.x < 64) {
        atomicAdd(&sum1[threadIdx.x], ssum[threadIdx.x]);
        atomicAdd(&sq1[threadIdx.x], ssq[threadIdx.x]);
    }
}

// Per-channel BN affine: a = g*rsqrt(var+eps), b = beta - mean*a
__global__ void bn_affine(const float* __restrict__ sum, const float* __restrict__ sq,
                          const float* __restrict__ gamma, const float* __restrict__ beta,
                          float* __restrict__ a, float* __restrict__ b, float invN)
{
    int c = threadIdx.x;
    if (c < 64) {
        float m   = sum[c] * invN;
        float var = fmaxf(sq[c] * invN - m * m, 0.f);
        float s   = gamma[c] * rsqrtf(var + 1e-5f);
        a[c] = s;
        b[c] = fmaf(-m, s, beta[c]);
    }
}

// ---------------------------------------------------------------------------
// FFN1: t = relu(h1 @ W1 + bb1), h1 = BN1 affine applied on the A-tile load.
// 8 waves = 8 column tiles of the 128-wide output.
// ---------------------------------------------------------------------------
__global__ void ffn1_kernel(const float* __restrict__ x1,
                            const float* __restrict__ a1, const float* __restrict__ b1,
                            const float* __restrict__ W1, const float* __restrict__ bb1,
                            float* __restrict__ t, int N)
{
    __shared__ __align__(16) float As[16 * 68];
    const int row0 = blockIdx.x * 16;
    const bool full = (row0 + 16 <= N);
    for (int i = threadIdx.x; i < 16 * 64; i += blockDim.x) {
        int r = i >> 6, c = i & 63;
        int gr = row0 + r;
        As[r * 68 + c] = (full || gr < N) ? fmaf(x1[(size_t)gr * 64 + c], a1[c], b1[c]) : 0.f;
    }
    __syncthreads();

    const int lane = threadIdx.x & 31;
    const int wv   = threadIdx.x >> 5;                  // 0..7
    const int hi   = lane >> 4;
    const int lo   = lane & 15;
    const int n0   = wv * 16;

    v8f c = {};
#pragma unroll
    for (int k0 = 0; k0 < 64; k0 += 4) {
        int kb = k0 + 2 * hi;
        v2f a = { As[lo * 68 + kb], As[lo * 68 + kb + 1] };
        v2f b = { W1[kb * 128 + n0 + lo], W1[(kb + 1) * 128 + n0 + lo] };
        WMMA_F32(c, a, b);
    }
    float bias = bb1[n0 + lo];
    if (full) {
#pragma unroll
        for (int r = 0; r < 8; ++r)
            t[(size_t)(row0 + r + 8 * hi) * 128 + n0 + lo] = fmaxf(c[r] + bias, 0.f);
    } else {
#pragma unroll
        for (int r = 0; r < 8; ++r) {
            int gr = row0 + r + 8 * hi;
            if (gr < N) t[(size_t)gr * 128 + n0 + lo] = fmaxf(c[r] + bias, 0.f);
        }
    }
}

// ---------------------------------------------------------------------------
// FFN2: x2 = h1 + (t @ W2 + bb2); fused per-channel stats for BN2.
// ---------------------------------------------------------------------------
__global__ void ffn2_kernel(const float* __restrict__ t, const float* __restrict__ x1,
                            const float* __restrict__ a1, const float* __restrict__ b1,
                            const float* __restrict__ W2, const float* __restrict__ bb2,
                            float* __restrict__ x2,
                            float* __restrict__ sum2, float* __restrict__ sq2, int N)
{
    __shared__ __align__(16) float As[16 * 132];        // 528B row stride: aligned + conflict-free
    __shared__ float ssum[64], ssq[64];
    const int row0 = blockIdx.x * 16;
    const bool full = (row0 + 16 <= N);
    if (threadIdx.x < 64) { ssum[threadIdx.x] = 0.f; ssq[threadIdx.x] = 0.f; }
    if (full) {
        for (int i = threadIdx.x; i < 512; i += blockDim.x) {
            int r = i >> 5, c = (i & 31) * 4;
            *(float4*)(As + r * 132 + c) = *(const float4*)(t + (size_t)(row0 + r) * 128 + c);
        }
    } else {
        for (int i = threadIdx.x; i < 16 * 128; i += blockDim.x) {
            int r = i >> 7, c = i & 127;
            int gr = row0 + r;
            As[r * 132 + c] = (gr < N) ? t[(size_t)gr * 128 + c] : 0.f;
        }
    }
    __syncthreads();

    const int lane = threadIdx.x & 31;
    const int wv   = threadIdx.x >> 5;                  // 0..3
    const int hi   = lane >> 4;
    const int lo   = lane & 15;
    const int n0   = wv * 16;

    v8f c = {};
#pragma unroll
    for (int k0 = 0; k0 < 128; k0 += 4) {
        int kb = k0 + 2 * hi;
        v2f a = { As[lo * 132 + kb], As[lo * 132 + kb + 1] };
        v2f b = { W2[kb * 64 + n0 + lo], W2[(kb + 1) * 64 + n0 + lo] };
        WMMA_F32(c, a, b);
    }

    const int col  = n0 + lo;
    const float bias = bb2[col];
    const float ac = a1[col], bc = b1[col];
    float ls = 0.f, lq = 0.f;
    if (full) {
#pragma unroll
        for (int r = 0; r < 8; ++r) {
            size_t gr = (size_t)(row0 + r + 8 * hi);
            float h1v = fmaf(x1[gr * 64 + col], ac, bc);
            float v   = h1v + c[r] + bias;
            x2[gr * 64 + col] = v;
            ls += v;
            lq += v * v;
        }
    } else {
#pragma unroll
        for (int r = 0; r < 8; ++r) {
            int gr = row0 + r + 8 * hi;
            if (gr < N) {
                float h1v = fmaf(x1[(size_t)gr * 64 + col], ac, bc);
                float v   = h1v + c[r] + bias;
                x2[(size_t)gr * 64 + col] = v;
                ls += v;
                lq += v * v;
            }
        }
    }
    atomicAdd(&ssum[col], ls);
    atomicAdd(&ssq[col], lq);
    __syncthreads();
    if (threadIdx.x < 64) {
        atomicAdd(&sum2[threadIdx.x], ssum[threadIdx.x]);
        atomicAdd(&sq2[threadIdx.x], ssq[threadIdx.x]);
    }
}

__global__ void bn_apply(const float* __restrict__ x, const float* __restrict__ a,
                         const float* __restrict__ b, float* __restrict__ out, int total4)
{
    const int cg = (threadIdx.x & 15) * 4;
    float4 av = *(const float4*)(a + cg);
    float4 bv = *(const float4*)(b + cg);
    const int stride = gridDim.x * blockDim.x;          // multiple of 16
    for (int i = blockIdx.x * blockDim.x + threadIdx.x; i < total4; i += stride) {
        float4 xv = *(const float4*)(x + (size_t)i * 4);
        float4 o;
        o.x = fmaf(xv.x, av.x, bv.x);
        o.y = fmaf(xv.y, av.y, bv.y);
        o.z = fmaf(xv.z, av.z, bv.z);
        o.w = fmaf(xv.w, av.w, bv.w);
        *(float4*)(out + (size_t)i * 4) = o;
    }
}

// ---------------------------------------------------------------------------
extern "C" void kernel_launch(void* const* d_in, const int* in_sizes, int n_in,
                              void* d_out, int out_size, void* d_ws, size_t ws_size,
                              hipStream_t stream)
{
    const float* h   = (const float*)d_in[0];
    const int*   src = (const int*)  d_in[1];
    const int*   dst = (const int*)  d_in[2];
    const float* Wq  = (const float*)d_in[3];
    const float* bq  = (const float*)d_in[4];
    const float* Wk  = (const float*)d_in[5];
    const float* bk  = (const float*)d_in[6];
    const float* Wv  = (const float*)d_in[7];
    const float* bv  = (const float*)d_in[8];
    const float* g1  = (const float*)d_in[9];
    const float* b1  = (const float*)d_in[10];
    const float* W1  = (const float*)d_in[11];
    const float* bb1 = (const float*)d_in[12];
    const float* W2  = (const float*)d_in[13];
    const float* bb2 = (const float*)d_in[14];
    const float* g2  = (const float*)d_in[15];
    const float* b2  = (const float*)d_in[16];
    float* out = (float*)d_out;

    const int N = in_sizes[0] / 64;   // 50000
    const int E = in_sizes[1];        // 800000
    const size_t ND = (size_t)N * 64;

    float* ws = (float*)d_ws;
    float* Q     = ws;
    float* K     = Q  + ND;
    float* V     = K  + ND;
    float* wV    = V  + ND;
    float* z     = wV + ND;                 // N*8
    float* stats = z  + (size_t)N * 8;
    float* sum1 = stats;       float* sq1 = stats + 64;
    float* sum2 = stats + 128; float* sq2 = stats + 192;
    float* a1s  = stats + 256; float* b1s = stats + 320;
    float* a2s  = stats + 384; float* b2s = stats + 448;
    // dead-buffer aliases after the edge phase:
    float* t  = Q;    // N*128 spans Q|K
    float* x1 = V;    // N*64
    float* x2 = wV;   // N*64

    // Zero accumulators (wV, z, sum/sq) — graph-capture-safe async memset.
    hipMemsetAsync(wV, 0, (ND + (size_t)N * 8 + 256) * sizeof(float), stream);

    const int rowBlocks = (N + 15) / 16;    // 3125
    qkv_kernel<<<rowBlocks, 128, 0, stream>>>(h, Wq, bq, Wk, bk, Wv, bv, Q, K, V, N);
    edge_attn<<<2048, 256, 0, stream>>>(Q, K, V, src, dst, wV, z, E);
    attn_resid_stats<<<512, 256, 0, stream>>>(h, wV, z, x1, sum1, sq1, N);
    bn_affine<<<1, 64, 0, stream>>>(sum1, sq1, g1, b1, a1s, b1s, 1.0f / (float)N);
    ffn1_kernel<<<rowBlocks, 256, 0, stream>>>(x1, a1s, b1s, W1, bb1, t, N);
    ffn2_kernel<<<rowBlocks, 128, 0, stream>>>(t, x1, a1s, b1s, W2, bb2, x2, sum2, sq2, N);
    bn_affine<<<1, 64, 0, stream>>>(sum2, sq2, g2, b2, a2s, b2s, 1.0f / (float)N);
    bn_apply<<<512, 256, 0, stream>>>(x2, a2s, b2s, out, N * 16);
}